// Attn_loc_47863115547246
// MI455X (gfx1250) — compile-verified
//
#include <hip/hip_runtime.h>
#include <hip/hip_bf16.h>
#include <stdint.h>

#define N_POI     10000
#define SEQ_LEN   2048
#define STATE_LEN 1024
#define THREADS   256
#define WAVES     (THREADS / 32)
#define PER_THR   (SEQ_LEN / THREADS)     // 8 elements per thread
#define ROW_B128  ((N_POI * 4) / 16)      // 2500 16-byte chunks per matrix row
#define FULL_IT   (ROW_B128 / THREADS)    // 9 unconditional issue rounds

__global__ __launch_bounds__(THREADS)
void attn_loc_softmax_kernel(const int*   __restrict__ his,
                             const int*   __restrict__ cur,
                             const float* __restrict__ mat,
                             float*       __restrict__ out)
{
    // 40000 B row staging buffer + tiny reduction scratch (8 blocks fit in a 320KB WGP)
    __shared__ __align__(16) float rowS[N_POI];
    __shared__ float redMax[WAVES];
    __shared__ float redSum[WAVES];

    const int tid = threadIdx.x;
    const int row = blockIdx.x;            // 0..1023

    const int c = cur[row];                // uniform -> scalar load
    const float* __restrict__ rowG = mat + (size_t)c * N_POI;  // 16B aligned (c*40000)

    // LDS byte offset: low 32 bits of the generic (addrspace 3->0 cast) pointer
    const uint32_t ldsBase = (uint32_t)(uintptr_t)(&rowS[0]);

    // ---- CDNA5 async bulk copy: global -> LDS, tracked by ASYNCcnt ----
    // 2500 chunks = 9 full rounds of 256 lanes + one partial (196 lanes, EXEC-masked).
    // Fully unrolled: 9 unconditional async issues + 1 predicated, no loop overhead.
#pragma unroll
    for (int k = 0; k < FULL_IT + 1; ++k) {
        const int i = tid + k * THREADS;
        if (i < ROW_B128) {
            uint32_t     lds_off = ldsBase + (uint32_t)i * 16u;
            const float* g       = rowG + i * 4;
            asm volatile("global_load_async_to_lds_b128 %0, %1, off"
                         :
                         : "v"(lds_off), "v"(g)
                         : "memory");
        }
    }

    // ---- his indices: issue BEFORE waiting on the DMA so their latency is hidden
    // under the 40KB row copy (LOADcnt is independent of ASYNCcnt) ----
    const int j0 = tid * PER_THR;
    const int4* hv = (const int4*)(his + j0);
    int4 ha = hv[0];
    int4 hb = hv[1];

    asm volatile("s_wait_asynccnt 0" ::: "memory");
    __syncthreads();   // make all waves' async writes visible block-wide

    int h[PER_THR];
    h[0] = ha.x; h[1] = ha.y; h[2] = ha.z; h[3] = ha.w;
    h[4] = hb.x; h[5] = hb.y; h[6] = hb.z; h[7] = hb.w;

    // ---- gather from LDS, energies, single-pass softmax ----
    float e[PER_THR];
    float m = -3.402823466e+38f;
#pragma unroll
    for (int k = 0; k < PER_THR; ++k) {
        float d = rowS[h[k]];
        float v = (d != 0.0f) ? (1.0f / d) : 1e-6f;
        e[k] = v;
        m = fmaxf(m, v);
    }

    // wave32 max reduction (warpSize == 32 on gfx1250)
#pragma unroll
    for (int off = 16; off > 0; off >>= 1)
        m = fmaxf(m, __shfl_xor(m, off, 32));
    const int wave = tid >> 5;
    if ((tid & 31) == 0) redMax[wave] = m;
    __syncthreads();
    m = redMax[0];
#pragma unroll
    for (int w = 1; w < WAVES; ++w) m = fmaxf(m, redMax[w]);

    // exp + sum (values stay in registers: one-pass softmax, no re-read)
    float s = 0.0f;
#pragma unroll
    for (int k = 0; k < PER_THR; ++k) {
        e[k] = __expf(e[k] - m);
        s += e[k];
    }
#pragma unroll
    for (int off = 16; off > 0; off >>= 1)
        s += __shfl_xor(s, off, 32);
    if ((tid & 31) == 0) redSum[wave] = s;
    __syncthreads();
    s = 0.0f;
#pragma unroll
    for (int w = 0; w < WAVES; ++w) s += redSum[w];

    const float inv = 1.0f / s;
    float4 o0 = make_float4(e[0] * inv, e[1] * inv, e[2] * inv, e[3] * inv);
    float4 o1 = make_float4(e[4] * inv, e[5] * inv, e[6] * inv, e[7] * inv);
    float4* op = (float4*)(out + (size_t)row * SEQ_LEN + j0);
    op[0] = o0;
    op[1] = o1;
}

extern "C" void kernel_launch(void* const* d_in, const int* in_sizes, int n_in,
                              void* d_out, int out_size, void* d_ws, size_t ws_size,
                              hipStream_t stream) {
    (void)in_sizes; (void)n_in; (void)out_size; (void)d_ws; (void)ws_size;
    const int*   his = (const int*)d_in[0];
    const int*   cur = (const int*)d_in[1];
    const float* mat = (const float*)d_in[2];
    float*       out = (float*)d_out;

    dim3 grid(STATE_LEN);   // one block per output row
    dim3 block(THREADS);    // 8 wave32 waves
    attn_loc_softmax_kernel<<<grid, block, 0, stream>>>(his, cur, mat, out);
}